// TokenSemantics_31275951849694
// MI455X (gfx1250) — compile-verified
//
#include <hip/hip_runtime.h>
#include <hip/hip_bf16.h>
#include <math.h>

// ---------------------------------------------------------------------------
// Sizes (fixed by the reference)
// ---------------------------------------------------------------------------
#define NT_N 100000
#define NP_N 50000
#define E_N  800000
#define H_D  128

typedef float v8f __attribute__((ext_vector_type(8)));
typedef float v2f __attribute__((ext_vector_type(2)));

// ---------------------------------------------------------------------------
// fp32 WMMA GEMM:  Y[N,128] = X[N,128] @ W[128,128] + b   (+ optional leaky)
// grid.x = N/16 (N divisible by 16); block = 256 threads = 8 wave32s.
// Wave w computes the 16x16 tile (rows 16*blockIdx.x.., cols 16*w..).
// A-matrix 16x4 f32 layout: lanes 0-15 hold M=lane, VGPR0=K0,VGPR1=K1;
// lanes 16-31 hold M=lane-16, VGPR0=K2,VGPR1=K3.  B mirrored with lane=N.
// C/D 16x16 f32: VGPR v, lanes 0-15 -> M=v, lanes 16-31 -> M=v+8.
// ---------------------------------------------------------------------------
__global__ void gemm128_wmma(const float* __restrict__ X,
                             const float* __restrict__ W,
                             const float* __restrict__ b,
                             float* __restrict__ Y,
                             int act) {
  const int wave = threadIdx.x >> 5;
  const int lane = threadIdx.x & 31;
  const int hf   = lane >> 4;   // which half of the wave
  const int lid  = lane & 15;
  const size_t rowBase = (size_t)blockIdx.x * 16;
  const int colBase = wave * 16;

  const float* Xr = X + (rowBase + lid) * H_D;
  v8f c = {};
#pragma unroll 8
  for (int k0 = 0; k0 < H_D; k0 += 4) {
    const int ka = k0 + 2 * hf;
    v2f a, bb;
    a[0]  = Xr[ka];
    a[1]  = Xr[ka + 1];
    bb[0] = W[(size_t)ka * H_D + colBase + lid];
    bb[1] = W[(size_t)(ka + 1) * H_D + colBase + lid];
    c = __builtin_amdgcn_wmma_f32_16x16x4_f32(false, a, false, bb,
                                              (short)0, c, false, false);
  }
  const float bias = b[colBase + lid];
#pragma unroll
  for (int v = 0; v < 8; ++v) {
    float val = c[v] + bias;
    if (act) val = (val > 0.0f) ? val : 0.01f * val;
    Y[(rowBase + v + 8 * hf) * H_D + colBase + lid] = val;
  }
}

// ---------------------------------------------------------------------------
// h_t[i,j] = x_token[i] * W_emb[j] + b_emb[j]
// ---------------------------------------------------------------------------
__global__ void embed_kernel(const float* __restrict__ xt,
                             const float* __restrict__ W,
                             const float* __restrict__ b,
                             float* __restrict__ Hout) {
  int i = blockIdx.x * blockDim.x + threadIdx.x;
  if (i >= NT_N * H_D) return;
  int n = i >> 7, j = i & 127;
  Hout[i] = xt[n] * W[j] + b[j];
}

__global__ void fill_kernel(float* __restrict__ p, int n, float v) {
  int i = blockIdx.x * blockDim.x + threadIdx.x;
  if (i < n) p[i] = v;
}

// float atomic max via signed/unsigned int trick (works with -inf init)
__device__ inline void atomicMaxFloat(float* addr, float val) {
  if (val >= 0.0f)
    atomicMax((int*)addr, __float_as_int(val));
  else
    atomicMin((unsigned int*)addr, __float_as_uint(val));
}

// wave-per-edge: scores[e] = dot(q[dst[e]], k[src[e]]) / sqrt(128); seg-max.
__global__ void edge_score_kernel(const float* __restrict__ Q,
                                  const float* __restrict__ K,
                                  const int* __restrict__ src,
                                  const int* __restrict__ dst,
                                  float* __restrict__ SC,
                                  float* __restrict__ M) {
  const int wave = threadIdx.x >> 5, lane = threadIdx.x & 31;
  const int e = blockIdx.x * 8 + wave;
  if (e >= E_N) return;
  const int s = src[e], d = dst[e];
  const float4 q4 = ((const float4*)(Q + (size_t)d * H_D))[lane];
  const float4 k4 = ((const float4*)(K + (size_t)s * H_D))[lane];
  float p = q4.x * k4.x + q4.y * k4.y + q4.z * k4.z + q4.w * k4.w;
#pragma unroll
  for (int off = 16; off >= 1; off >>= 1) p += __shfl_xor(p, off, 32);
  if (lane == 0) {
    const float sc = p * 0.08838834764831845f;  // 1/sqrt(128)
    SC[e] = sc;
    atomicMaxFloat(&M[d], sc);
  }
}

// thread-per-edge: SC[e] <- exp(SC[e]-M[dst]); denom[dst] += SC[e]
__global__ void edge_expsum_kernel(float* __restrict__ SC,
                                   const float* __restrict__ M,
                                   float* __restrict__ DEN,
                                   const int* __restrict__ dst) {
  const int e = blockIdx.x * blockDim.x + threadIdx.x;
  if (e >= E_N) return;
  const int d = dst[e];
  const float ev = __expf(SC[e] - M[d]);
  SC[e] = ev;
  atomicAdd(&DEN[d], ev);
}

// wave-per-edge: AGG[dst] += alpha * v[src]
__global__ void edge_scatter_kernel(const float* __restrict__ SC,
                                    const float* __restrict__ DEN,
                                    const float* __restrict__ V,
                                    const int* __restrict__ src,
                                    const int* __restrict__ dst,
                                    float* __restrict__ AGG) {
  const int wave = threadIdx.x >> 5, lane = threadIdx.x & 31;
  const int e = blockIdx.x * 8 + wave;
  if (e >= E_N) return;
  const int s = src[e], d = dst[e];
  const float alpha = SC[e] / (DEN[d] + 1e-16f);
  const float4 v4 = ((const float4*)(V + (size_t)s * H_D))[lane];
  float* o = AGG + (size_t)d * H_D + lane * 4;
  atomicAdd(o + 0, alpha * v4.x);
  atomicAdd(o + 1, alpha * v4.y);
  atomicAdd(o + 2, alpha * v4.z);
  atomicAdd(o + 3, alpha * v4.w);
}

// wave-per-node: beta = sigmoid([out,skip,out-skip] @ Wb); gated mix (+leaky)
__global__ void gate_kernel(const float* __restrict__ AGG,
                            const float* __restrict__ S,
                            const float* __restrict__ Wb,
                            float* __restrict__ dest,
                            int N, int act) {
  const int wave = threadIdx.x >> 5, lane = threadIdx.x & 31;
  const int n = blockIdx.x * 8 + wave;
  if (n >= N) return;
  const float4 o4 = ((const float4*)(AGG + (size_t)n * H_D))[lane];
  const float4 s4 = ((const float4*)(S + (size_t)n * H_D))[lane];
  const float4 w0 = ((const float4*)(Wb))[lane];
  const float4 w1 = ((const float4*)(Wb + H_D))[lane];
  const float4 w2 = ((const float4*)(Wb + 2 * H_D))[lane];
  float p = o4.x * w0.x + o4.y * w0.y + o4.z * w0.z + o4.w * w0.w +
            s4.x * w1.x + s4.y * w1.y + s4.z * w1.z + s4.w * w1.w +
            (o4.x - s4.x) * w2.x + (o4.y - s4.y) * w2.y +
            (o4.z - s4.z) * w2.z + (o4.w - s4.w) * w2.w;
#pragma unroll
  for (int off = 16; off >= 1; off >>= 1) p += __shfl_xor(p, off, 32);
  const float beta = 1.0f / (1.0f + __expf(-p));
  float4 r;
  r.x = beta * s4.x + (1.0f - beta) * o4.x;
  r.y = beta * s4.y + (1.0f - beta) * o4.y;
  r.z = beta * s4.z + (1.0f - beta) * o4.z;
  r.w = beta * s4.w + (1.0f - beta) * o4.w;
  if (act) {
    r.x = (r.x > 0.0f) ? r.x : 0.01f * r.x;
    r.y = (r.y > 0.0f) ? r.y : 0.01f * r.y;
    r.z = (r.z > 0.0f) ? r.z : 0.01f * r.z;
    r.w = (r.w > 0.0f) ? r.w : 0.01f * r.w;
  }
  ((float4*)(dest + (size_t)n * H_D))[lane] = r;
}

// ---------------------------------------------------------------------------
// Host orchestration
// ---------------------------------------------------------------------------
extern "C" void kernel_launch(void* const* d_in, const int* in_sizes, int n_in,
                              void* d_out, int out_size, void* d_ws, size_t ws_size,
                              hipStream_t stream) {
  (void)in_sizes; (void)n_in; (void)out_size;
  const float* x_token  = (const float*)d_in[0];
  const float* x_phrase = (const float*)d_in[1];
  const float* W_emb = (const float*)d_in[2];
  const float* b_emb = (const float*)d_in[3];
  const float* Wq_tp = (const float*)d_in[4];  const float* bq_tp = (const float*)d_in[5];
  const float* Wk_tp = (const float*)d_in[6];  const float* bk_tp = (const float*)d_in[7];
  const float* Wv_tp = (const float*)d_in[8];  const float* bv_tp = (const float*)d_in[9];
  const float* Ws_tp = (const float*)d_in[10]; const float* bs_tp = (const float*)d_in[11];
  const float* Wb_tp = (const float*)d_in[12];
  const float* Wq_pt = (const float*)d_in[13]; const float* bq_pt = (const float*)d_in[14];
  const float* Wk_pt = (const float*)d_in[15]; const float* bk_pt = (const float*)d_in[16];
  const float* Wv_pt = (const float*)d_in[17]; const float* bv_pt = (const float*)d_in[18];
  const float* Ws_pt = (const float*)d_in[19]; const float* bs_pt = (const float*)d_in[20];
  const float* Wb_pt = (const float*)d_in[21];
  const float* W_head = (const float*)d_in[22]; const float* b_head = (const float*)d_in[23];
  const int* src_tp = (const int*)d_in[24]; const int* dst_tp = (const int*)d_in[25];
  const int* src_pt = (const int*)d_in[26]; const int* dst_pt = (const int*)d_in[27];

  // Workspace layout (floats). Lifetime-based aliasing:
  //   Hbuf: h_t, later overwritten by t1.   Qb: q, later the softmax aggregate.
  //   Kb:   k, later t2.
  float* ws   = (float*)d_ws;
  float* Hbuf = ws;
  float* P1   = Hbuf + (size_t)NT_N * H_D;
  float* Qb   = P1   + (size_t)NP_N * H_D;
  float* Kb   = Qb   + (size_t)NT_N * H_D;
  float* Vb   = Kb   + (size_t)NT_N * H_D;
  float* Sb   = Vb   + (size_t)NT_N * H_D;
  float* SC   = Sb   + (size_t)NT_N * H_D;
  float* Mb   = SC   + (size_t)E_N;
  float* DENb = Mb   + (size_t)NT_N;
  const size_t need = (size_t)((DENb + NT_N) - ws) * sizeof(float);
  if (ws_size < need) return;

  const dim3 blk(256);
  const int edgeBlocks = (E_N + 7) / 8;

  auto run_conv = [&](const float* xsrc, int Nsrc, const float* xdst, int Ndst,
                      const float* Wq, const float* bq, const float* Wk, const float* bk,
                      const float* Wv, const float* bv, const float* Wsk, const float* bsk,
                      const float* Wb, const int* src, const int* dst,
                      float* dest, int act) {
    gemm128_wmma<<<dim3(Ndst / 16), blk, 0, stream>>>(xdst, Wq,  bq,  Qb, 0);
    gemm128_wmma<<<dim3(Nsrc / 16), blk, 0, stream>>>(xsrc, Wk,  bk,  Kb, 0);
    gemm128_wmma<<<dim3(Nsrc / 16), blk, 0, stream>>>(xsrc, Wv,  bv,  Vb, 0);
    gemm128_wmma<<<dim3(Ndst / 16), blk, 0, stream>>>(xdst, Wsk, bsk, Sb, 0);
    fill_kernel<<<(Ndst + 255) / 256, blk, 0, stream>>>(Mb,   Ndst, -INFINITY);
    fill_kernel<<<(Ndst + 255) / 256, blk, 0, stream>>>(DENb, Ndst, 0.0f);
    edge_score_kernel<<<edgeBlocks, blk, 0, stream>>>(Qb, Kb, src, dst, SC, Mb);
    edge_expsum_kernel<<<(E_N + 255) / 256, blk, 0, stream>>>(SC, Mb, DENb, dst);
    // Qb is dead now -> reuse as the aggregate buffer
    fill_kernel<<<(Ndst * H_D + 255) / 256, blk, 0, stream>>>(Qb, Ndst * H_D, 0.0f);
    edge_scatter_kernel<<<edgeBlocks, blk, 0, stream>>>(SC, DENb, Vb, src, dst, Qb);
    gate_kernel<<<(Ndst + 7) / 8, blk, 0, stream>>>(Qb, Sb, Wb, dest, Ndst, act);
  };

  // h_t = x_token @ W_emb + b_emb
  embed_kernel<<<(NT_N * H_D + 255) / 256, blk, 0, stream>>>(x_token, W_emb, b_emb, Hbuf);

  // conv1 tp: tokens -> phrases, p1 (leaky relu fused)
  run_conv(Hbuf, NT_N, x_phrase, NP_N,
           Wq_tp, bq_tp, Wk_tp, bk_tp, Wv_tp, bv_tp, Ws_tp, bs_tp, Wb_tp,
           src_tp, dst_tp, P1, 1);

  // conv1 pt: phrases -> tokens, t1 overwrites Hbuf (h_t dead after its GEMMs)
  run_conv(x_phrase, NP_N, Hbuf, NT_N,
           Wq_pt, bq_pt, Wk_pt, bk_pt, Wv_pt, bv_pt, Ws_pt, bs_pt, Wb_pt,
           src_pt, dst_pt, Hbuf, 1);

  // conv2 pt (shared params): src = p1, dst = t1; t2 lands in Kb (k dead by then)
  run_conv(P1, NP_N, Hbuf, NT_N,
           Wq_pt, bq_pt, Wk_pt, bk_pt, Wv_pt, bv_pt, Ws_pt, bs_pt, Wb_pt,
           src_pt, dst_pt, Kb, 0);

  // head: out = t2 @ W_head + b_head
  gemm128_wmma<<<dim3(NT_N / 16), blk, 0, stream>>>(Kb, W_head, b_head, (float*)d_out, 0);
}